// OskarLayer_17635135717859
// MI455X (gfx1250) — compile-verified
//
#include <hip/hip_runtime.h>
#include <hip/hip_bf16.h>

// Sizes fixed by the reference
#define BATCH 2
#define SEQ   384
#define HID   768
#define NHEAD 12
#define DHEAD 64
#define ROWS  (BATCH * SEQ)      // 768
#define L2E   1.4426950408889634f
#define NT    (SEQ / 32)         // 12 k-tiles per (b,h)

typedef float v2f __attribute__((ext_vector_type(2)));
typedef float v8f __attribute__((ext_vector_type(8)));

// D = A(16x4,f32) x B(4x16,f32) + C(16x16,f32)  -> v_wmma_f32_16x16x4_f32
__device__ __forceinline__ v8f wmma_f32_4(v2f a, v2f b, v8f c) {
    return __builtin_amdgcn_wmma_f32_16x16x4_f32(
        /*neg_a=*/false, a, /*neg_b=*/false, b,
        /*c_mod=*/(short)0, c, /*reuse_a=*/false, /*reuse_b=*/false);
}

// ---------------------------------------------------------------------------
// Tensor Data Mover: 2D f32 tile (rows x 64) global -> LDS with row padding.
// pad_interval code 5 => pad after every 64 DWORDs (one row); pad_amount code
// 1 => +2 DW (stride 66), code 3 => +4 DW (stride 68).
// Toolchain here exposes the 6-arg builtin (clang-23 / therock-10.0):
//   (u32x4 g0, i32x8 g1, i32x4 g2, i32x4 g3, i32x8 extra, i32 cpol)
// The extra 8-DWORD operand corresponds to the ISA's unused VADDR4 slot ->
// pass zeros.
// ---------------------------------------------------------------------------
#if defined(__HIP_DEVICE_COMPILE__) && \
    __has_builtin(__builtin_amdgcn_tensor_load_to_lds) && \
    __has_builtin(__builtin_amdgcn_s_wait_tensorcnt)
#define USE_TDM 1
typedef unsigned int u32x4 __attribute__((ext_vector_type(4)));
typedef int          i32x8 __attribute__((ext_vector_type(8)));
typedef int          i32x4 __attribute__((ext_vector_type(4)));

__device__ __forceinline__ void tdm_load_rows64(const float* gsrc, unsigned lds_off,
                                                int rows, int pad_code) {
    const unsigned long long ga = (unsigned long long)(uintptr_t)gsrc;
    u32x4 g0;
    g0[0] = 1u;                                         // count=1, user mode
    g0[1] = lds_off;                                    // LDS byte address
    g0[2] = (unsigned)ga;                               // global_addr[31:0]
    g0[3] = ((unsigned)(ga >> 32) & 0x01FFFFFFu) | 0x80000000u; // addr[56:32]|type=2
    i32x8 g1;
    g1[0] = (2 << 16) | (1 << 20) | (5 << 22) | (pad_code << 25); // 4B, pad en
    g1[1] = (64  << 16);                                // tensor_dim0 = 64
    g1[2] = (rows << 16);                               // tensor_dim1 = rows
    g1[3] = (64  << 16);                                // tile_dim0 = 64
    g1[4] = rows;                                       // tile_dim1 = rows
    g1[5] = 64;                                         // tensor_dim0_stride = 64
    g1[6] = 0;
    g1[7] = 0;
    i32x4 g2; g2[0] = 1; g2[1] = 1; g2[2] = 0; g2[3] = 0;   // dim2=dim3=1 (unused)
    i32x4 g3; g3[0] = 0; g3[1] = (1 << 16); g3[2] = 0; g3[3] = 0; // dim4=1
    i32x8 g4; g4[0] = 0; g4[1] = 0; g4[2] = 0; g4[3] = 0;   // VADDR4 slot: unused
    g4[4] = 0; g4[5] = 0; g4[6] = 0; g4[7] = 0;
    __builtin_amdgcn_tensor_load_to_lds(g0, g1, g2, g3, g4, 0);
}
#else
#define USE_TDM 0
#endif

// ---------------------------------------------------------------------------
// Kernel 1: fused QKV projection.  out = x @ W.T + b, written as [B,NH,S,DH].
// One wave computes a 16x32 strip for q, k AND v: one A fragment -> 6 WMMAs.
// Wave-tiles: 48 row-tiles x 24 col-pair-tiles = 1152; 4 waves/block -> 288.
// ---------------------------------------------------------------------------
__global__ __launch_bounds__(128) void qkv_wmma_kernel(
    const float* __restrict__ x,
    const float* __restrict__ Wq, const float* __restrict__ bq,
    const float* __restrict__ Wk, const float* __restrict__ bk,
    const float* __restrict__ Wv, const float* __restrict__ bv,
    float* __restrict__ qo, float* __restrict__ ko, float* __restrict__ vo)
{
    const int wave = threadIdx.x >> 5;
    const int lane = threadIdx.x & 31;
    const int tile = blockIdx.x * 4 + wave;          // 0..1151
    const int tm = tile / (HID / 32);                // row tile (16 rows)
    const int tp = tile % (HID / 32);                // col-pair tile (32 cols)
    const int row0 = tm * 16;
    const int col0 = tp * 32;

    const int m    = lane & 15;                      // A row / B col / D col
    const int half = lane >> 4;

    const float* xrow = x  + (size_t)(row0 + m) * HID;        // A: x[row][k]
    const float* wq0  = Wq + (size_t)(col0      + m) * HID;   // B[k][n] = W[n][k]
    const float* wq1  = Wq + (size_t)(col0 + 16 + m) * HID;
    const float* wk0  = Wk + (size_t)(col0      + m) * HID;
    const float* wk1  = Wk + (size_t)(col0 + 16 + m) * HID;
    const float* wv0  = Wv + (size_t)(col0      + m) * HID;
    const float* wv1  = Wv + (size_t)(col0 + 16 + m) * HID;

    v8f cq0 = {}; v8f cq1 = {};
    v8f ck0 = {}; v8f ck1 = {};
    v8f cv0 = {}; v8f cv1 = {};

    #pragma unroll 4
    for (int kb = 0; kb < HID; kb += 4) {
        const int ko_ = kb + 2 * half;               // even -> 8B aligned
        v2f a = *(const v2f*)(xrow + ko_);
        cq0 = wmma_f32_4(a, *(const v2f*)(wq0 + ko_), cq0);
        cq1 = wmma_f32_4(a, *(const v2f*)(wq1 + ko_), cq1);
        ck0 = wmma_f32_4(a, *(const v2f*)(wk0 + ko_), ck0);
        ck1 = wmma_f32_4(a, *(const v2f*)(wk1 + ko_), ck1);
        cv0 = wmma_f32_4(a, *(const v2f*)(wv0 + ko_), cv0);
        cv1 = wmma_f32_4(a, *(const v2f*)(wv1 + ko_), cv1);
    }

    #pragma unroll
    for (int c = 0; c < 2; ++c) {
        const int col  = col0 + c * 16 + m;
        const int head = col >> 6;
        const int d    = col & 63;
        const float biasq = bq[col], biask = bk[col], biasv = bv[col];
        const v8f aq = c ? cq1 : cq0;
        const v8f ak = c ? ck1 : ck0;
        const v8f av = c ? cv1 : cv0;
        #pragma unroll
        for (int r = 0; r < 8; ++r) {
            const int row = row0 + r + 8 * half;     // global token row
            const int b   = row / SEQ;               // tile never spans batches
            const int s   = row % SEQ;
            const size_t o = (((size_t)b * NHEAD + head) * SEQ + s) * DHEAD + d;
            qo[o] = aq[r] + biasq;
            ko[o] = ak[r] + biask;
            vo[o] = av[r] + biasv;
        }
    }
}

// ---------------------------------------------------------------------------
// Kernel 2: fused attention for one (b, h, 16-row i-tile).
//   scores_ij = -rsqrt(sqrt(sum_d 1/(q_id - k_jd)^2)), softmax over j,
//   ctx(16x64) = probs(16x384) @ v(384x64) via WMMA (A from LDS).
// k tiles double-buffered through LDS via the Tensor Data Mover.
// Grid: B*NH*(S/16) = 576 blocks, 128 threads (4 waves).
// ---------------------------------------------------------------------------
#define SQS 66    // sQ row stride: 8B aligned, conflict-free (pad 2 = TDM code 1)
#define SKS 68    // sK row stride: 16B aligned, conflict-free (pad 4 = TDM code 3)
#define SPS 386   // sP row stride: 8B aligned & conflict-free for WMMA A reads

__global__ __launch_bounds__(128) void attn_kernel(
    const float* __restrict__ qg, const float* __restrict__ kg,
    const float* __restrict__ vg, float* __restrict__ ctx)
{
    __shared__ alignas(16) float sQ[16][SQS];
    __shared__ alignas(16) float sK[2][32][SKS];
    __shared__ alignas(16) float sP[16][SPS];
    __shared__ float pred[16][8];
    __shared__ float sinv[16];

    const int tid = threadIdx.x;
    const int blk = blockIdx.x;                  // (b*NH + h)*24 + it
    const int it  = blk % (SEQ / 16);
    const int bh  = blk / (SEQ / 16);
    const int i0  = it * 16;
    const size_t base = (size_t)bh * SEQ * DHEAD;

    // ---- stage q tile + first k tile ----
#if USE_TDM
    if (tid < 32) {
        tdm_load_rows64(qg + base + (size_t)i0 * DHEAD,
                        (unsigned)(uintptr_t)&sQ[0][0], 16, /*pad 2 DW*/ 1);
        tdm_load_rows64(kg + base,
                        (unsigned)(uintptr_t)&sK[0][0][0], 32, /*pad 4 DW*/ 3);
    }
#else
    for (int t = tid; t < 16 * DHEAD; t += 128)
        sQ[t >> 6][t & 63] = qg[base + (size_t)(i0 + (t >> 6)) * DHEAD + (t & 63)];
    for (int t = tid; t < 32 * DHEAD; t += 128)
        sK[0][t >> 6][t & 63] = kg[base + (size_t)(t >> 6) * DHEAD + (t & 63)];
#endif

    // ---- phase 1: scores; k tiles double-buffered ----
    const int i  = tid & 15;                     // score row for this thread
    const int j0 = tid >> 4;                     // 0..7; handles j0, j0+8, +16, +24
    for (int jt = 0; jt < NT; ++jt) {
        const int cur = jt & 1;
#if USE_TDM
        if (tid < 32 && jt + 1 < NT)             // prefetch next tile (in-order TDM)
            tdm_load_rows64(kg + base + (size_t)(jt + 1) * 32 * DHEAD,
                            (unsigned)(uintptr_t)&sK[1 - cur][0][0], 32, 3);
        if (jt + 1 < NT) __builtin_amdgcn_s_wait_tensorcnt(1);  // tile jt done
        else             __builtin_amdgcn_s_wait_tensorcnt(0);
#else
        if (jt + 1 < NT)
            for (int t = tid; t < 32 * DHEAD; t += 128)
                sK[1 - cur][t >> 6][t & 63] =
                    kg[base + (size_t)((jt + 1) * 32 + (t >> 6)) * DHEAD + (t & 63)];
#endif
        __syncthreads();

        float a0 = 0.0f, a1 = 0.0f, a2 = 0.0f, a3 = 0.0f;
        #pragma unroll
        for (int d = 0; d < DHEAD; d += 2) {
            const v2f qv = *(const v2f*)&sQ[i][d];
            const v2f k0 = *(const v2f*)&sK[cur][j0     ][d];
            const v2f k1 = *(const v2f*)&sK[cur][j0 +  8][d];
            const v2f k2 = *(const v2f*)&sK[cur][j0 + 16][d];
            const v2f k3 = *(const v2f*)&sK[cur][j0 + 24][d];
            float df;
            df = qv.x - k0.x; a0 += __builtin_amdgcn_rcpf(df * df);
            df = qv.y - k0.y; a0 += __builtin_amdgcn_rcpf(df * df);
            df = qv.x - k1.x; a1 += __builtin_amdgcn_rcpf(df * df);
            df = qv.y - k1.y; a1 += __builtin_amdgcn_rcpf(df * df);
            df = qv.x - k2.x; a2 += __builtin_amdgcn_rcpf(df * df);
            df = qv.y - k2.y; a2 += __builtin_amdgcn_rcpf(df * df);
            df = qv.x - k3.x; a3 += __builtin_amdgcn_rcpf(df * df);
            df = qv.y - k3.y; a3 += __builtin_amdgcn_rcpf(df * df);
        }
        const int jb = jt * 32 + j0;
        sP[i][jb     ] = -__builtin_amdgcn_rsqf(__builtin_amdgcn_sqrtf(a0));
        sP[i][jb +  8] = -__builtin_amdgcn_rsqf(__builtin_amdgcn_sqrtf(a1));
        sP[i][jb + 16] = -__builtin_amdgcn_rsqf(__builtin_amdgcn_sqrtf(a2));
        sP[i][jb + 24] = -__builtin_amdgcn_rsqf(__builtin_amdgcn_sqrtf(a3));
        __syncthreads();
    }

    // ---- phase 2: softmax over j (row r handled by 8 threads g=0..7) ----
    const int r = tid & 15;
    const int g = tid >> 4;
    float pm = -__builtin_inff();
    for (int j = g; j < SEQ; j += 8) pm = fmaxf(pm, sP[r][j]);
    pred[r][g] = pm;
    __syncthreads();
    float mrow = pred[r][0];
    #pragma unroll
    for (int t = 1; t < 8; ++t) mrow = fmaxf(mrow, pred[r][t]);
    __syncthreads();
    float ps = 0.0f;
    for (int j = g; j < SEQ; j += 8) {
        const float e = __builtin_amdgcn_exp2f((sP[r][j] - mrow) * L2E);
        sP[r][j] = e;
        ps += e;
    }
    pred[r][g] = ps;
    __syncthreads();
    float tot = 0.0f;
    #pragma unroll
    for (int t = 0; t < 8; ++t) tot += pred[r][t];
    if (g == 0) sinv[r] = __builtin_amdgcn_rcpf(tot);   // fold 1/sum into store
    __syncthreads();

    // ---- phase 3: ctx = probs @ v via WMMA f32 16x16x4 ----
    const int wv   = tid >> 5;                   // n-tile: dh cols wv*16..+15
    const int lane = tid & 31;
    const int n    = lane & 15;
    const int half = lane >> 4;
    const int col  = wv * 16 + n;
    const float* vb = vg + base;

    v8f c = {};
    #pragma unroll 4
    for (int kb = 0; kb < SEQ; kb += 4) {
        const int kk = kb + 2 * half;
        v2f a = *(const v2f*)&sP[n][kk];         // A row m == n (lane&15)
        v2f b;
        b.x = vb[(size_t)(kk + 0) * DHEAD + col];
        b.y = vb[(size_t)(kk + 1) * DHEAD + col];
        c = wmma_f32_4(a, b, c);
    }

    #pragma unroll
    for (int rr = 0; rr < 8; ++rr) {
        const int M = rr + 8 * half;
        ctx[base + (size_t)(i0 + M) * DHEAD + col] = c[rr] * sinv[M];
    }
}

// ---------------------------------------------------------------------------
// Kernel 3: y = x + ctx (head transpose folded in), then LayerNorm.
// One block per token row; wave32 shfl reductions.
// ---------------------------------------------------------------------------
__global__ __launch_bounds__(256) void ln_kernel(
    const float* __restrict__ x, const float* __restrict__ ctx,
    const float* __restrict__ gamma, const float* __restrict__ beta,
    float* __restrict__ out)
{
    __shared__ float sred[8];
    const int row = blockIdx.x;                  // b*SEQ + s
    const int b   = row / SEQ;
    const int s   = row % SEQ;
    const int tid = threadIdx.x;

    float y[3];
    #pragma unroll
    for (int u = 0; u < 3; ++u) {
        const int h = tid + u * 256;
        y[u] = x[(size_t)row * HID + h] +
               ctx[(((size_t)b * NHEAD + (h >> 6)) * SEQ + s) * DHEAD + (h & 63)];
    }

    float p = y[0] + y[1] + y[2];
    #pragma unroll
    for (int off = 16; off > 0; off >>= 1) p += __shfl_xor(p, off, 32);
    if ((tid & 31) == 0) sred[tid >> 5] = p;
    __syncthreads();
    float tot = 0.0f;
    #pragma unroll
    for (int i = 0; i < 8; ++i) tot += sred[i];
    const float mu = tot * (1.0f / HID);
    __syncthreads();

    float q = 0.0f;
    #pragma unroll
    for (int u = 0; u < 3; ++u) { const float d = y[u] - mu; q += d * d; }
    #pragma unroll
    for (int off = 16; off > 0; off >>= 1) q += __shfl_xor(q, off, 32);
    if ((tid & 31) == 0) sred[tid >> 5] = q;
    __syncthreads();
    float v2 = 0.0f;
    #pragma unroll
    for (int i = 0; i < 8; ++i) v2 += sred[i];
    const float inv = __builtin_amdgcn_rsqf(v2 * (1.0f / HID) + 1e-12f);

    #pragma unroll
    for (int u = 0; u < 3; ++u) {
        const int h = tid + u * 256;
        out[(size_t)row * HID + h] = (y[u] - mu) * inv * gamma[h] + beta[h];
    }
}

// ---------------------------------------------------------------------------
extern "C" void kernel_launch(void* const* d_in, const int* in_sizes, int n_in,
                              void* d_out, int out_size, void* d_ws, size_t ws_size,
                              hipStream_t stream)
{
    (void)in_sizes; (void)n_in; (void)out_size; (void)ws_size;
    const float* x  = (const float*)d_in[0];
    const float* Wq = (const float*)d_in[1];
    const float* bq = (const float*)d_in[2];
    const float* Wk = (const float*)d_in[3];
    const float* bk = (const float*)d_in[4];
    const float* Wv = (const float*)d_in[5];
    const float* bv = (const float*)d_in[6];
    const float* lg = (const float*)d_in[7];
    const float* lb = (const float*)d_in[8];
    float* out = (float*)d_out;

    float* ws = (float*)d_ws;
    const size_t N = (size_t)ROWS * HID;         // 589824 floats per tensor
    float* qw = ws;
    float* kw = ws + N;
    float* vw = ws + 2 * N;
    float* cw = ws + 3 * N;

    qkv_wmma_kernel<<<dim3((HID / 16) * (HID / 32) / 4), dim3(128), 0, stream>>>(
        x, Wq, bq, Wk, bk, Wv, bv, qw, kw, vw);
    attn_kernel<<<dim3(BATCH * NHEAD * (SEQ / 16)), dim3(128), 0, stream>>>(
        qw, kw, vw, cw);
    ln_kernel<<<dim3(ROWS), dim3(256), 0, stream>>>(x, cw, lg, lb, out);
}